// PointLoss_eval_3925600108611
// MI455X (gfx1250) — compile-verified
//
#include <hip/hip_runtime.h>
#include <hip/hip_bf16.h>
#include <math.h>

// Chamfer distance on MI455X (gfx1250), wave32 — two-pass WMMA formulation.
//
// d2[q,r] = |q|^2 + |r|^2 - 2 q.r as a K=4 Gram update with
// V_WMMA_F32_16X16X4_F32 (CDNA5's only f32 WMMA shape; K=4 fits D=3 + norm):
//   A row q  = ( x, y, z, |q|^2 )     (16x4 A: 2 VGPRs; lanes 0-15 K={0,1},
//                                      lanes 16-31 K={2,3})
//   B col r  = (-2x,-2y,-2z, 1 )      (4x16 B: mirrored layout)
//   C[q,r]   = -2 q.r + |q|^2 ;  d2 = C + |r|^2 (per-lane add, lane == col)
//
// Each wave owns one 16-row tile of the "query" set and loops the "ref" set
// in 16-col tiles, accumulating min(d2) purely in per-lane registers (C
// layout: row = v + 8*half, col = lane&15). One 4-step shfl_xor reduction at
// the end. No atomics, no LDS waits, no divergence in the hot loop. Pass 1:
// query=array2 (dist1 mins); Pass 2: query=array1 (dist2 mins). Compute is
// ~2.4 GFLOP total and all data is L2-resident (400KB << 192MB), so doubling
// WMMA work to remove per-iteration sync is the right trade on MI455X.

typedef float v2f __attribute__((ext_vector_type(2)));
typedef float v8f __attribute__((ext_vector_type(8)));

#define WAVES_PER_BLOCK 8
#define ROWS_PER_BLOCK (WAVES_PER_BLOCK * 16)

__global__ __launch_bounds__(256) void pl_min_dist(
    const float* __restrict__ Q,   // [B, NQ, 3] query points (A rows, owned)
    const float* __restrict__ R,   // [B, NR, 3] ref points   (B cols, looped)
    float* __restrict__ outmin,    // [B, NQ] : min over r of d2(q, r)
    int B, int NQ, int NR)
{
    const int qChunks = NQ / ROWS_PER_BLOCK;
    const int b      = blockIdx.x / qChunks;
    const int qchunk = blockIdx.x % qChunks;
    const int wave = threadIdx.x >> 5;
    const int lane = threadIdx.x & 31;
    const int half = lane >> 4;     // K/M half selector per ISA layouts
    const int l16  = lane & 15;
    const int qtile = qchunk * ROWS_PER_BLOCK + wave * 16;

    // ---- A fragment (fixed for this wave's 16 query rows) ----
    const float* pq = Q + (size_t)(b * NQ + qtile + l16) * 3;
    const float qx = pq[0], qy = pq[1], qz = pq[2];
    const float r2q = qx * qx + qy * qy + qz * qz;
    v2f af;
    if (half == 0) { af.x = qx; af.y = qy; }
    else           { af.x = qz; af.y = r2q; }

    float rmin[8];
#pragma unroll
    for (int v = 0; v < 8; ++v) rmin[v] = __int_as_float(0x7F800000);

#pragma unroll 2
    for (int rt = 0; rt < NR; rt += 16) {
        // ---- B fragment for this 16-col ref tile ----
        const float* pr = R + (size_t)(b * NR + rt + l16) * 3;
        const float rx = pr[0], ry = pr[1], rz = pr[2];
        const float r2r = rx * rx + ry * ry + rz * rz;
        v2f bf;
        if (half == 0) { bf.x = -2.0f * rx; bf.y = -2.0f * ry; }
        else           { bf.x = -2.0f * rz; bf.y = 1.0f; }

        v8f c = {};
        c = __builtin_amdgcn_wmma_f32_16x16x4_f32(
                /*neg_a=*/false, af, /*neg_b=*/false, bf,
                /*c_mod=*/(short)0, c, /*reuse_a=*/false, /*reuse_b=*/false);

        // d2 = C + |r|^2 ; accumulate per-lane running min (no cross-lane
        // work in the loop — the reduction is deferred to the end).
#pragma unroll
        for (int v = 0; v < 8; ++v)
            rmin[v] = fminf(rmin[v], c[v] + r2r);
    }

    // ---- Final min reduction across the 16 lanes of each half ----
#pragma unroll
    for (int v = 0; v < 8; ++v) {
        float r = rmin[v];
        r = fminf(r, __shfl_xor(r, 1, 32));
        r = fminf(r, __shfl_xor(r, 2, 32));
        r = fminf(r, __shfl_xor(r, 4, 32));
        r = fminf(r, __shfl_xor(r, 8, 32));
        if (l16 == 0)   // lane 0 writes rows v+0..7, lane 16 rows v+8..15
            outmin[(size_t)b * NQ + qtile + half * 8 + v] = fmaxf(r, 0.0f);
    }
}

__global__ __launch_bounds__(256) void pl_reduce(
    const float* __restrict__ min1,   // [B,M] per-array2-point mins
    const float* __restrict__ min2,   // [B,N] per-array1-point mins
    float* __restrict__ out,          // [3*B]: dist_all | dist1 | dist2
    int B, int N, int M)
{
    __shared__ float s1[256];
    __shared__ float s2[256];
    const int b = blockIdx.x;
    float a1 = 0.0f, a2 = 0.0f;
    for (int i = threadIdx.x; i < M; i += 256) a1 += min1[(size_t)b * M + i];
    for (int i = threadIdx.x; i < N; i += 256) a2 += min2[(size_t)b * N + i];
    s1[threadIdx.x] = a1;
    s2[threadIdx.x] = a2;
    __syncthreads();
    for (int s = 128; s > 0; s >>= 1) {
        if (threadIdx.x < s) {
            s1[threadIdx.x] += s1[threadIdx.x + s];
            s2[threadIdx.x] += s2[threadIdx.x + s];
        }
        __syncthreads();
    }
    if (threadIdx.x == 0) {
        const float d1 = s1[0] / (float)M;   // mean over array2 points
        const float d2 = s2[0] / (float)N;   // mean over array1 points
        out[b]         = d1 + d2;            // dist_all
        out[B + b]     = d1;                 // dist1
        out[2 * B + b] = d2;                 // dist2
    }
}

extern "C" void kernel_launch(void* const* d_in, const int* in_sizes, int n_in,
                              void* d_out, int out_size, void* d_ws, size_t ws_size,
                              hipStream_t stream) {
    const float* array1 = (const float*)d_in[0];  // [B,N,3]
    const float* array2 = (const float*)d_in[1];  // [B,M,3]
    const int B = 8;
    const int N = in_sizes[0] / (B * 3);
    const int M = in_sizes[1] / (B * 3);

    float* min1 = (float*)d_ws;                   // [B,M]: min over array1
    float* min2 = min1 + (size_t)B * M;           // [B,N]: min over array2

    // Pass 1: each array2 point vs all array1 points -> dist1 mins
    pl_min_dist<<<B * (M / ROWS_PER_BLOCK), 256, 0, stream>>>(
        array2, array1, min1, B, M, N);

    // Pass 2: each array1 point vs all array2 points -> dist2 mins
    pl_min_dist<<<B * (N / ROWS_PER_BLOCK), 256, 0, stream>>>(
        array1, array2, min2, B, N, M);

    pl_reduce<<<B, 256, 0, stream>>>(min1, min2, (float*)d_out, B, N, M);
}